// Gemma4VisionPooler_33655363731987
// MI455X (gfx1250) — compile-verified
//
#include <hip/hip_runtime.h>
#include <hip/hip_bf16.h>
#include <math.h>

// Problem constants (from reference)
#define BB   8
#define SS   4096
#define HH   1152
#define LL   256
#define KK   4          // pooling kernel k (sqrt(S/L))

#define NT   288        // 9 wave32s per block
#define CH   15         // ceil(4096/288): flat values per thread (register-cached)
#define TPW  8          // column tiles (of 16) per wave: 9*8*16 = 1152

typedef float v2f __attribute__((ext_vector_type(2)));
typedef float v8f __attribute__((ext_vector_type(8)));

// ---------------- Kernel 1: per-batch max_x -> bin stride multiplier -------
__global__ void k_maxx(const int* __restrict__ pos, int* __restrict__ mult) {
    __shared__ int red[256];
    const int b = blockIdx.x;
    const int t = threadIdx.x;
    int m = 0;
    for (int s = t; s < SS; s += 256) {
        int px = pos[((size_t)b * SS + s) * 2];
        if (px < 0) px = 0;                       // safe = max(p,0)
        m = max(m, px);
    }
    red[t] = m;
    __syncthreads();
    for (int off = 128; off > 0; off >>= 1) {
        if (t < off) red[t] = max(red[t], red[t + off]);
        __syncthreads();
    }
    if (t == 0) mult[b] = (red[0] + 1) / KK;      // (max_x) // k
}

// ---------------- Kernel 2: flat bin id per token --------------------------
__global__ void k_flat(const int* __restrict__ pos, const int* __restrict__ mult,
                       int* __restrict__ flat) {
    const int tid = blockIdx.x * blockDim.x + threadIdx.x;
    if (tid >= BB * SS) return;
    const int b  = tid >> 12;                     // / SS
    const int px = pos[2 * (size_t)tid + 0];
    const int py = pos[2 * (size_t)tid + 1];
    const bool pad = (px == -1) && (py == -1);    // w = 0 -> contributes nothing
    const int sx = px < 0 ? 0 : px;
    const int sy = py < 0 ? 0 : py;
    int f = (sx / KK) + mult[b] * (sy / KK);
    if (pad || (unsigned)f >= (unsigned)LL) f = -1; // match segment_sum OOB drop
    flat[tid] = f;
}

// ---------------- Kernel 3: per (batch,bin) gather + WMMA column-sum -------
// Per-bin reduction = ones(16x4) x B(4x16) via V_WMMA_F32_16X16X4_F32,
// 4 k-steps per group of 16 gathered rows. A = all-ones makes the result
// invariant to the exact K-slot permutation of the B VGPR layout.
__global__ __launch_bounds__(NT)
void k_pool(const float* __restrict__ X, const int* __restrict__ flat,
            float* __restrict__ out, float* __restrict__ maskout) {
    __shared__ int list[SS];                      // ordered matching token ids
    __shared__ int cnts[NT + 1];

    const int l = blockIdx.x;
    const int b = blockIdx.y;
    const int t = threadIdx.x;
    const int* __restrict__ fb = flat + (size_t)b * SS;

    // --- Phase A: deterministic ordered compaction (register-cached flat) ---
    const int s0 = t * CH;
    int myf[CH];
#pragma unroll
    for (int i = 0; i < CH; ++i)
        myf[i] = (s0 + i < SS) ? fb[s0 + i] : -2; // -2 never matches any bin

    int c = 0;
#pragma unroll
    for (int i = 0; i < CH; ++i) c += (myf[i] == l);
    cnts[t] = c;
    __syncthreads();
    if (t == 0) {                                 // tiny serial exclusive scan
        int run = 0;
        for (int i = 0; i < NT; ++i) { int v = cnts[i]; cnts[i] = run; run += v; }
        cnts[NT] = run;
    }
    __syncthreads();
    int off = cnts[t];
#pragma unroll
    for (int i = 0; i < CH; ++i)
        if (myf[i] == l) list[off++] = s0 + i;    // sorted by s -> deterministic
    __syncthreads();
    const int cnt = cnts[NT];

    // --- Phase B: WMMA column sums --------------------------------------
    const int lane  = t & 31;                     // wave32
    const int wave  = t >> 5;
    const int lo    = lane & 15;                  // N (column within tile)
    const int hi    = lane >> 4;                  // half-wave -> K slot group
    const int tile0 = wave * TPW;

    v8f acc[TPW] = {};                            // zero accumulators
    v2f a; a[0] = 1.0f; a[1] = 1.0f;              // ones A-matrix (2 VGPRs)

    const float* __restrict__ Xb = X + (size_t)b * SS * HH;
    const int full = cnt & ~15;                   // whole groups of 16 rows

    // Fast path: full groups -> unconditional non-temporal loads, no selects.
    for (int j0 = 0; j0 < full; j0 += 16) {       // uniform loop: EXEC stays full
#pragma unroll
        for (int kk = 0; kk < 4; ++kk) {
            const int ja = j0 + kk * 4 + 2 * hi;  // per-lane row slots
            const int ra = list[ja];
            const int rb = list[ja + 1];
            const float* pa = Xb + (size_t)ra * HH + lo;
            const float* pb = Xb + (size_t)rb * HH + lo;
            float xa[TPW], xb[TPW];
#pragma unroll
            for (int i = 0; i < TPW; ++i) {       // batch all 16 loads first
                xa[i] = __builtin_nontemporal_load(&pa[(tile0 + i) * 16]);
                xb[i] = __builtin_nontemporal_load(&pb[(tile0 + i) * 16]);
            }
#pragma unroll
            for (int i = 0; i < TPW; ++i) {       // then 8 independent wmma chains
                v2f bv; bv[0] = xa[i]; bv[1] = xb[i];
                acc[i] = __builtin_amdgcn_wmma_f32_16x16x4_f32(
                    false, a, false, bv, (short)0, acc[i], false, false);
            }
        }
    }

    // Tail: one partial group. Loads stay unconditional (clamped index),
    // zero-padding done with selects -> v_cndmask, no exec branches.
    if (full < cnt) {
#pragma unroll
        for (int kk = 0; kk < 4; ++kk) {
            const int ja = full + kk * 4 + 2 * hi;
            const int jb = ja + 1;
            const int ra = list[min(ja, cnt - 1)];
            const int rb = list[min(jb, cnt - 1)];
            const bool va = (ja < cnt);
            const bool vb = (jb < cnt);
            const float* pa = Xb + (size_t)ra * HH + lo;
            const float* pb = Xb + (size_t)rb * HH + lo;
            float xa[TPW], xb[TPW];
#pragma unroll
            for (int i = 0; i < TPW; ++i) {
                xa[i] = __builtin_nontemporal_load(&pa[(tile0 + i) * 16]);
                xb[i] = __builtin_nontemporal_load(&pb[(tile0 + i) * 16]);
            }
#pragma unroll
            for (int i = 0; i < TPW; ++i) {
                v2f bv;
                bv[0] = va ? xa[i] : 0.0f;
                bv[1] = vb ? xb[i] : 0.0f;
                acc[i] = __builtin_amdgcn_wmma_f32_16x16x4_f32(
                    false, a, false, bv, (short)0, acc[i], false, false);
            }
        }
    }

    // pooled = sum/16 * sqrt(1152)  ->  scale = sqrt(1152)/16 = 1.5*sqrt(2)
    const float scale = 2.1213203435596424f;
    float* __restrict__ ob = out + ((size_t)b * LL + l) * HH;
    if (lane < 16) {                              // D row M=0 = VGPR0, lanes 0-15
#pragma unroll
        for (int i = 0; i < TPW; ++i)
            ob[(tile0 + i) * 16 + lo] = acc[i][0] * scale;
    }
    if (t == 0) maskout[(size_t)b * LL + l] = (cnt > 0) ? 1.0f : 0.0f;
}

// ---------------------------------------------------------------------------
extern "C" void kernel_launch(void* const* d_in, const int* in_sizes, int n_in,
                              void* d_out, int out_size, void* d_ws, size_t ws_size,
                              hipStream_t stream) {
    const float* hidden = (const float*)d_in[0];   // [8,4096,1152] f32
    const int*   pos    = (const int*)d_in[1];     // [8,4096,2] i32

    int* mult = (int*)d_ws;                        // [8]
    int* flat = mult + 8;                          // [8*4096]

    float* pooled = (float*)d_out;                 // [8,256,1152]
    float* maskf  = pooled + (size_t)BB * LL * HH; // [8,256] as 0/1 floats

    k_maxx<<<BB, 256, 0, stream>>>(pos, mult);
    k_flat<<<(BB * SS + 255) / 256, 256, 0, stream>>>(pos, mult, flat);
    k_pool<<<dim3(LL, BB), NT, 0, stream>>>(hidden, flat, pooled, maskf);
}